// LocalSelfAttentionBase_39470749450363
// MI455X (gfx1250) — compile-verified
//
#include <hip/hip_runtime.h>

// ---- problem constants (match reference config) ----
#define KVOL   27          // 3^3 kernel offsets
#define HALFK  1
#define SSTR   258         // COORD_RANGE + 2*HALF
#define FEAT   32
#define HASH_BITS 22
#define HASH_SIZE (1u << HASH_BITS)
#define HASH_MASK (HASH_SIZE - 1u)
#define SCAN_THREADS 256
#define SCAN_ITEMS   16
#define SCAN_TILE    (SCAN_THREADS * SCAN_ITEMS)   // 4096
#define TILES_PER_WAVE 2                            // 32 entries / wave in scatter

typedef float v2f __attribute__((ext_vector_type(2)));
typedef float v8f __attribute__((ext_vector_type(8)));

// ---------------------------------------------------------------- init
__global__ void k_init_hash(int* __restrict__ hk, int* __restrict__ hf) {
    unsigned stride = gridDim.x * blockDim.x;
    for (unsigned i = blockIdx.x * blockDim.x + threadIdx.x; i < HASH_SIZE; i += stride) {
        hk[i] = -1;
        hf[i] = 0x7fffffff;
    }
}

// vectorized constant fill (b128 stores on the 345MB out_feats zero-init)
__global__ void k_fill(float* __restrict__ p, long long n, float val) {
    long long stride = (long long)gridDim.x * blockDim.x;
    long long tid = blockIdx.x * (long long)blockDim.x + threadIdx.x;
    if ((((unsigned long long)p) & 15ull) == 0ull) {
        long long n4 = n >> 2;
        float4* p4 = (float4*)p;
        float4 v4 = make_float4(val, val, val, val);
        for (long long i = tid; i < n4; i += stride) p4[i] = v4;
        for (long long i = (n4 << 2) + tid; i < n; i += stride) p[i] = val;
    } else {
        for (long long i = tid; i < n; i += stride) p[i] = val;
    }
}

// ---------------------------------------------------------------- hash insert
// deterministic: slot keyed by enc value (CAS), first-occurrence via atomicMin
__global__ void k_insert(const int* __restrict__ si, int* __restrict__ hk,
                         int* __restrict__ hf, int* __restrict__ eslot,
                         int N, int NK) {
    int e = blockIdx.x * blockDim.x + threadIdx.x;
    if (e >= NK) return;
    int i = e / KVOL;
    int k = e - i * KVOL;
    __builtin_prefetch(&si[i + N], 0, 1);           // global_prefetch_b8
    int b = si[i];
    // shifted neighbor coords: coord + offset + HALF == coord + k-component
    int x = si[N     + i] + (k / 9);
    int y = si[2 * N + i] + ((k / 3) % 3);
    int z = si[3 * N + i] + (k % 3);
    int enc = ((b * SSTR + x) * SSTR + y) * SSTR + z;   // fits int32

    unsigned h = ((unsigned)enc * 2654435761u) >> (32 - HASH_BITS);
    for (;;) {
        int prev = atomicCAS(&hk[h], -1, enc);
        if (prev == -1 || prev == enc) break;
        h = (h + 1) & HASH_MASK;
    }
    atomicMin(&hf[h], e);
    eslot[e] = (int)h;
}

// ---------------------------------------------------------------- first-occurrence flags
__global__ void k_flag(const int* __restrict__ hf, const int* __restrict__ eslot,
                       int* __restrict__ scan, int NK) {
    int e = blockIdx.x * blockDim.x + threadIdx.x;
    if (e >= NK) return;
    scan[e] = (hf[eslot[e]] == e) ? 1 : 0;
}

// ---------------------------------------------------------------- scan stage 1 (per 4096 tile)
__global__ void k_scan_block(int* __restrict__ scan, int* __restrict__ bsum, int n) {
    __shared__ int sh[SCAN_THREADS];
    int t = threadIdx.x;
    int base = blockIdx.x * SCAN_TILE + t * SCAN_ITEMS;
    int vals[SCAN_ITEMS];
    int s = 0;
#pragma unroll
    for (int j = 0; j < SCAN_ITEMS; ++j) {
        int idx = base + j;
        int v = (idx < n) ? scan[idx] : 0;
        s += v;
        vals[j] = s;                       // inclusive within thread
    }
    sh[t] = s;
    __syncthreads();
    for (int off = 1; off < SCAN_THREADS; off <<= 1) {
        int x = (t >= off) ? sh[t - off] : 0;
        __syncthreads();
        sh[t] += x;
        __syncthreads();
    }
    int excl = sh[t] - s;                  // exclusive prefix of thread totals
#pragma unroll
    for (int j = 0; j < SCAN_ITEMS; ++j) {
        int idx = base + j;
        if (idx < n) scan[idx] = vals[j] + excl;
    }
    if (t == SCAN_THREADS - 1) bsum[blockIdx.x] = sh[t];
}

// ---------------------------------------------------------------- scan stage 2 (<=1024 block sums)
__global__ void k_scan_top(int* __restrict__ bsum, int nb) {
    __shared__ int sh[1024];
    int t = threadIdx.x;
    int v = (t < nb) ? bsum[t] : 0;
    sh[t] = v;
    __syncthreads();
    for (int off = 1; off < 1024; off <<= 1) {
        int x = (t >= off) ? sh[t - off] : 0;
        __syncthreads();
        sh[t] += x;
        __syncthreads();
    }
    if (t < nb) bsum[t] = sh[t] - v;       // exclusive
}

// ---------------------------------------------------------------- scan stage 3 (add back)
__global__ void k_scan_add(int* __restrict__ scan, const int* __restrict__ bsum, int n) {
    int e = blockIdx.x * blockDim.x + threadIdx.x;
    if (e >= n) return;
    scan[e] += bsum[e >> 12];              // SCAN_TILE == 4096
}

// ---------------------------------------------------------------- kernel_map / out_key / out_idx
__global__ void k_finalize(const int* __restrict__ si, const int* __restrict__ hf,
                           const int* __restrict__ eslot, const int* __restrict__ scan,
                           int* __restrict__ oidx, float* __restrict__ km,
                           float* __restrict__ okey, int N, int NK) {
    int e = blockIdx.x * blockDim.x + threadIdx.x;
    if (e >= NK) return;
    int slot = eslot[e];
    int f    = hf[slot];
    int oid  = scan[f] - 1;                // first-seen rank
    oidx[e]  = oid;
    int i = e / KVOL;
    int k = e - i * KVOL;
    long long e3 = 3LL * e;
    km[e3 + 0] = (float)i;
    km[e3 + 1] = (float)oid;
    km[e3 + 2] = (float)k;
    if (f == e) {                          // first occurrence writes the decoded key
        int x = si[N     + i] + (k / 9)       - HALFK;
        int y = si[2 * N + i] + ((k / 3) % 3) - HALFK;
        int z = si[3 * N + i] + (k % 3)       - HALFK;
        long long o3 = 3LL * oid;
        okey[o3 + 0] = (float)x;
        okey[o3 + 1] = (float)y;
        okey[o3 + 2] = (float)z;
    }
}

// ---------------------------------------------------------------- WMMA gather + atomic scatter
// One wave handles TILES_PER_WAVE x 16 consecutive entries; each 16-entry tile
// spans <=2 distinct source voxels. D(16x16) = A(16x4 indicator: entry->voxel)
// x B(4x16: voxel feature half-rows): the matrix unit broadcasts 2 loaded rows
// into 16 per-entry rows (exact in f32), then lanes scatter with clause-batched
// global_atomic_add_f32. B is loaded UNCONDITIONALLY by all 32 lanes: high-half
// lanes feed B rows 2/3, which are multiplied by A columns 2/3 == 0, so no
// exec-mask branching is needed around the loads.
__global__ void k_scatter(const float* __restrict__ feat, const int* __restrict__ oidx,
                          float* __restrict__ ofeat, int N, int nWaves) {
    int wave = blockIdx.x * (blockDim.x >> 5) + (threadIdx.x >> 5);
    if (wave >= nWaves) return;            // wave-uniform: EXEC all-ones for WMMA
    int lane = threadIdx.x & 31;
    bool lowHalf = lane < 16;
    int col   = lane & 15;
    int rbase = lowHalf ? 0 : 8;           // D VGPR j holds rows j and j+8
    long long base = (long long)wave * (16 * TILES_PER_WAVE);
    __builtin_prefetch(&oidx[base], 0, 1);

#pragma unroll
    for (int t = 0; t < TILES_PER_WAVE; ++t) {
        long long e0 = base + 16 * t;
        int v0 = (int)(e0 / KVOL);
        int v1 = v0 + 1; if (v1 >= N) v1 = N - 1;   // unused rows have A==0

        // A 16x4 (f32, 2 VGPRs): VGPR0={K0|K2}, VGPR1={K1|K3} across lane halves
        int em = (int)e0 + col;
        int g  = (em / KVOL) - v0;                  // voxel slot 0/1
        v2f A;
        A.x = (lowHalf && g == 0) ? 1.0f : 0.0f;    // A[m][0] (A[m][2]==0)
        A.y = (lowHalf && g == 1) ? 1.0f : 0.0f;    // A[m][1] (A[m][3]==0)

        // out rows for this lane's 8 D registers (shared by both halves)
        int o[8];
#pragma unroll
        for (int j = 0; j < 8; ++j) o[j] = oidx[e0 + rbase + j];

#pragma unroll
        for (int h = 0; h < 2; ++h) {               // two 16-feature halves
            // unconditional broadcast loads; high lanes fill B rows 2/3 (x0 in A)
            v2f B;
            B.x = feat[(long long)v0 * FEAT + h * 16 + col];
            B.y = feat[(long long)v1 * FEAT + h * 16 + col];
            v8f C = {};
            v8f D = __builtin_amdgcn_wmma_f32_16x16x4_f32(
                /*neg_a=*/false, A, /*neg_b=*/false, B,
                /*c_mod=*/(short)0, C, /*reuse_a=*/false, /*reuse_b=*/false);
#pragma unroll
            for (int j = 0; j < 8; ++j)
                atomicAdd(&ofeat[(long long)o[j] * FEAT + h * 16 + col], D[j]);
        }
    }
}

// scalar tail for the <32 entries not covered by full WMMA tiles
__global__ void k_scatter_tail(const float* __restrict__ feat, const int* __restrict__ oidx,
                               float* __restrict__ ofeat, int eBase, int NK) {
    int tid = blockIdx.x * blockDim.x + threadIdx.x;
    int total = (NK - eBase) * FEAT;
    if (tid >= total) return;
    int e = eBase + tid / FEAT;
    int f = tid - (tid / FEAT) * FEAT;
    int i = e / KVOL;
    atomicAdd(&ofeat[(long long)oidx[e] * FEAT + f], feat[(long long)i * FEAT + f]);
}

// ---------------------------------------------------------------- launch
extern "C" void kernel_launch(void* const* d_in, const int* in_sizes, int n_in,
                              void* d_out, int out_size, void* d_ws, size_t ws_size,
                              hipStream_t stream) {
    const int*   si   = (const int*)d_in[0];     // sparse_indices [4, N]
    const float* feat = (const float*)d_in[1];   // features [N, 32]
    int N  = in_sizes[0] / 4;
    int NK = N * KVOL;

    float* out   = (float*)d_out;
    float* km    = out;                          // [NK,3]
    float* okey  = out + (size_t)NK * 3;         // [NK,3]
    float* ofeat = out + (size_t)NK * 6;         // [NK,32]

    // workspace carve
    char* w = (char*)d_ws;
    int* hk    = (int*)w; w += (size_t)HASH_SIZE * 4;
    int* hf    = (int*)w; w += (size_t)HASH_SIZE * 4;
    int* eslot = (int*)w; w += (size_t)NK * 4;
    int* scan  = (int*)w; w += (size_t)NK * 4;
    int* oidx  = (int*)w; w += (size_t)NK * 4;
    int* bsum  = (int*)w; w += 4096 * 4;

    int gE   = (NK + 255) / 256;
    int nblk = (NK + SCAN_TILE - 1) / SCAN_TILE;      // 660 for N=100k (<=1024 required)

    k_init_hash<<<512, 256, 0, stream>>>(hk, hf);
    k_fill<<<1024, 256, 0, stream>>>(okey, (long long)NK * 3, -1.0f);
    k_fill<<<4096, 256, 0, stream>>>(ofeat, (long long)NK * 32, 0.0f);
    k_insert<<<gE, 256, 0, stream>>>(si, hk, hf, eslot, N, NK);
    k_flag<<<gE, 256, 0, stream>>>(hf, eslot, scan, NK);
    k_scan_block<<<nblk, SCAN_THREADS, 0, stream>>>(scan, bsum, NK);
    k_scan_top<<<1, 1024, 0, stream>>>(bsum, nblk);
    k_scan_add<<<gE, 256, 0, stream>>>(scan, bsum, NK);
    k_finalize<<<gE, 256, 0, stream>>>(si, hf, eslot, scan, oidx, km, okey, N, NK);

    // full-tile WMMA scatter + scalar tail
    int entriesPerWave = 16 * TILES_PER_WAVE;          // 32
    int nWaves = NK / entriesPerWave;                  // full waves only
    if (nWaves > 0) {
        int sBlocks = (nWaves + 7) / 8;                // 8 waves / 256-thread block
        k_scatter<<<sBlocks, 256, 0, stream>>>(feat, oidx, ofeat, N, nWaves);
    }
    int eBase = nWaves * entriesPerWave;
    int tail  = NK - eBase;
    if (tail > 0) {
        int tThreads = tail * FEAT;
        k_scatter_tail<<<(tThreads + 255) / 256, 256, 0, stream>>>(feat, oidx, ofeat, eBase, NK);
    }
}